// Linear_12635793785535
// MI455X (gfx1250) — compile-verified
//
#include <hip/hip_runtime.h>

// OFT block-diagonal rotation for MI455X (gfx1250, wave32).
// Phase 1: build 64x64 rotation matrices via Cayley-Neumann (VALU, tiny),
//          stored K-PAIRED: rotp[r][k2][c] = {rot[2k2][c], rot[2k2+1][c]}
//          so the apply kernel's WMMA B-fragments are single b64 loads.
// Phase 2: out = x @ blockdiag(rot) using V_WMMA_F32_16X16X4_F32 (exact f32).
// Apply phase is HBM-bound (~268 MB @ 23.3 TB/s ~ 11.5 us); f32 WMMA is
// required because VALU f32 FMA can't reach the ~373 TF/s that the
// 16 FLOP/byte intensity demands at full HBM rate.

#define BLOCKSZ 64
#define R_BLOCKS 64
#define N_ELEM 2016           // 64*63/2
#define TOKENS 8192
#define IN_FEATURES 4096
#define S2 80                 // LDS row stride in float2 units (640 B):
                              // half-waves (K-pair rows differing by 1) hit
                              // disjoint 32-bank groups (2*S2 mod 64 == 32)

typedef float v2f __attribute__((ext_vector_type(2)));
typedef float v8f __attribute__((ext_vector_type(8)));

// ---------------------------------------------------------------------------
// Kernel A: rot[r] = I + 2Q + 2Q^2 + 2Q^3 + 2Q^4, Q skew-symmetric from
// packed upper-triangular weight[r]. One workgroup per block r.
// Output layout (per r): 2048 float2 = [k2=0..31][c=0..63] K-pairs.
// ---------------------------------------------------------------------------
__global__ __launch_bounds__(256) void build_rot_kernel(
    const float* __restrict__ w, float* __restrict__ rotp) {
  __shared__ float Q[BLOCKSZ][BLOCKSZ + 1];
  __shared__ float P[BLOCKSZ][BLOCKSZ + 1];
  const int r = blockIdx.x;
  const int tid = threadIdx.x;
  const float* wr = w + r * N_ELEM;

  int ii[16], jj[16];
  float acc[16];
#pragma unroll
  for (int t = 0; t < 16; ++t) {
    const int p = tid + t * 256;
    const int i = p >> 6, j = p & 63;
    ii[t] = i; jj[t] = j;
    float v = 0.0f;
    if (j > i)      v =  wr[i * (127 - i) / 2 + (j - i - 1)];
    else if (j < i) v = -wr[j * (127 - j) / 2 + (i - j - 1)];
    Q[i][j] = v;
    P[i][j] = v;
    acc[t] = (i == j ? 1.0f : 0.0f) + 2.0f * v;
  }
  __syncthreads();

  for (int term = 2; term < 5; ++term) {
    float tv[16];
#pragma unroll
    for (int t = 0; t < 16; ++t) {
      const int i = ii[t], j = jj[t];
      float s = 0.0f;
      for (int k = 0; k < BLOCKSZ; ++k) s += P[i][k] * Q[k][j];
      tv[t] = s;
    }
    __syncthreads();  // all reads of P done before overwrite
#pragma unroll
    for (int t = 0; t < 16; ++t) {
      P[ii[t]][jj[t]] = tv[t];
      acc[t] += 2.0f * tv[t];
    }
    __syncthreads();
  }

  // K-paired store: float index = (i>>1)*128 + j*2 + (i&1)
#pragma unroll
  for (int t = 0; t < 16; ++t) {
    const int i = ii[t], j = jj[t];
    rotp[r * (BLOCKSZ * BLOCKSZ) + (i >> 1) * 128 + (j << 1) + (i & 1)] =
        acc[t];
  }
}

// ---------------------------------------------------------------------------
// Kernel B: out[n, r*64 + c] = sum_k x[n, r*64 + k] * rot[r][k][c]
// 128 threads = 4 wave32. Each wave owns a 16-token x 64-col output stripe:
// K-loop of 16 steps x 4 column tiles of v_wmma_f32_16x16x4_f32.
//
// A-fragment (16x4 f32, ISA layout): lane L holds row M = L&15;
//   half = L>>4 selects K-pair: v0 = A[M, 4kk + 2*half], v1 = next K.
//   -> one global_load_b64 per K-step.
// B-fragment mirrors it: v0 = B[4kk + 2*half, N=L&15], v1 = next K row.
//   -> K-paired LDS layout makes this one aligned ds_load_b64 (no re-pairing
//      movs), conflict-free via the S2=80 stride.
// C/D (8 VGPRs): VGPR j -> row M = j + 8*half, col N = L&15.
// ---------------------------------------------------------------------------
__global__ __launch_bounds__(128) void apply_rot_kernel(
    const float* __restrict__ x, const float* __restrict__ rotp,
    float* __restrict__ out) {
  __shared__ v2f Bl[32 * S2];   // 20.0 KB
  const int r   = blockIdx.y;
  const int m0  = blockIdx.x * 64;
  const int tid = threadIdx.x;

  // Stage K-paired rot[r] into LDS: straight float2 copy with stride pad.
  const v2f* src = (const v2f*)(rotp + r * (BLOCKSZ * BLOCKSZ));
#pragma unroll
  for (int t = 0; t < 16; ++t) {
    const int e = tid + t * 128;            // 2048 float2 total
    Bl[(e >> 6) * S2 + (e & 63)] = src[e];  // b64 load + b64 ds store
  }
  __syncthreads();

  const int wave = tid >> 5;
  const int lane = tid & 31;
  const int n    = lane & 15;
  const int half = lane >> 4;

  // A pointer: this lane's token row + K-pair base for its half-wave.
  const float* xrow = x + (size_t)(m0 + wave * 16 + n) * IN_FEATURES
                        + r * BLOCKSZ + 2 * half;

  v8f acc[4] = {};

#pragma unroll
  for (int kk = 0; kk < 16; ++kk) {
    const v2f a = *(const v2f*)(xrow + kk * 4);          // global_load_b64
    const v2f* bp = Bl + (2 * kk + half) * S2 + n;       // K-pair row
#pragma unroll
    for (int nt = 0; nt < 4; ++nt) {
      const v2f b = bp[nt * 16];                         // ds_load_b64
      acc[nt] = __builtin_amdgcn_wmma_f32_16x16x4_f32(
          /*neg_a=*/false, a, /*neg_b=*/false, b,
          /*c_mod=*/(short)0, acc[nt],
          /*reuse_a=*/false, /*reuse_b=*/false);
    }
  }

  // Store: VGPR j -> token row m0 + wave*16 + j + 8*half.
  float* obase = out + (size_t)(m0 + wave * 16 + 8 * half) * IN_FEATURES
                     + r * BLOCKSZ + n;
#pragma unroll
  for (int j = 0; j < 8; ++j) {
#pragma unroll
    for (int nt = 0; nt < 4; ++nt)
      obase[(size_t)j * IN_FEATURES + nt * 16] = acc[nt][j];
  }
}

// ---------------------------------------------------------------------------
extern "C" void kernel_launch(void* const* d_in, const int* in_sizes, int n_in,
                              void* d_out, int out_size, void* d_ws,
                              size_t ws_size, hipStream_t stream) {
  (void)in_sizes; (void)n_in; (void)out_size; (void)ws_size;
  const float* x = (const float*)d_in[0];      // [8192, 4096] f32
  const float* w = (const float*)d_in[1];      // [64, 2016]  f32
  float* out = (float*)d_out;                  // [8192, 4096] f32
  float* rotp = (float*)d_ws;                  // [64][32][64] float2 = 1 MB

  build_rot_kernel<<<dim3(R_BLOCKS), dim3(256), 0, stream>>>(w, rotp);
  apply_rot_kernel<<<dim3(TOKENS / 64, R_BLOCKS), dim3(128), 0, stream>>>(
      x, rotp, out);
}